// CapsuleCoParall_61160334295615
// MI455X (gfx1250) — compile-verified
//
#include <hip/hip_runtime.h>

#define B_     16
#define NUM_   2
#define IN_N_  1152
#define IN_L_  16
#define OUT_N_ 64
#define OUT_L_ 32
#define COLS_  2048          // OUT_N*OUT_L
#define BSTR_  4096          // NUM*OUT_N*OUT_L  (batch stride in vraw/S)
#define ICHUNK_ 8
#define CHUNKS_ (IN_N_/ICHUNK_)   // 144
#define WPB_   16            // waves per block
#define CPW_   128           // columns per wave
#define TPW_   8             // 16-col tiles per wave
#define EPS_   1e-5f

typedef __attribute__((ext_vector_type(2))) float v2f;
typedef __attribute__((ext_vector_type(8))) float v8f;

// One 16x16 D tile: D = A(16x16) x B(16x16 slice of W^T) + bias.
// All 4 B-fragments are loaded first (one load clause, one wait), then 4 chained
// K=4 WMMAs. wlane = &W[n][i][col0 + (lane&15)][(lane>>4)*2] per-lane.
__device__ __forceinline__ v8f mm16(const float* __restrict__ wlane, const v2f* af, float bv) {
  v2f bf[4];
#pragma unroll
  for (int kk = 0; kk < 4; ++kk) bf[kk] = *(const v2f*)(wlane + kk * 4);
  v8f d = {bv, bv, bv, bv, bv, bv, bv, bv};
#pragma unroll
  for (int kk = 0; kk < 4; ++kk)
    d = __builtin_amdgcn_wmma_f32_16x16x4_f32(false, af[kk], false, bf[kk], (short)0, d,
                                              false, false);
  return d;
}

// One routing sweep. routing==0: cc comes from input c (first iteration).
// routing==1: logits = u_hat . S  (S = cumulative sum of squashed agreement vectors),
//             cc = softmax over out_n, accumulate cc*u_hat into vraw.
__global__ __launch_bounds__(512) void caps_pass(const float* __restrict__ u,
                                                 const float* __restrict__ w,
                                                 const float* __restrict__ bias,
                                                 const float* __restrict__ cvec,
                                                 float* __restrict__ vraw,
                                                 const float* __restrict__ S, int routing) {
  __shared__ float lg[B_][OUT_N_];
  __shared__ float cc[B_][OUT_N_];

  const int n    = blockIdx.x / CHUNKS_;
  const int i0   = (blockIdx.x % CHUNKS_) * ICHUNK_;
  const int tid  = threadIdx.x;
  const int lane = tid & 31;
  const int wv   = tid >> 5;        // wave 0..15, owns cols [wv*128, wv*128+128)
  const int half = lane >> 4;       // D rows: M = r + 8*half -> batch index
  const int l15  = lane & 15;       // D cols: N = l15 within tile

  // Per-lane, per-tile (o,v) index into [n][o][v] and bias value (u_hat includes bias).
  int   sb[TPW_];
  float bv[TPW_];
#pragma unroll
  for (int t = 0; t < TPW_; ++t) {
    int col = wv * CPW_ + t * 16 + l15;
    int o = col >> 5, vc = col & 31;
    sb[t] = (n * OUT_N_ + o) * OUT_L_ + vc;
    bv[t] = bias[sb[t]];
  }

  float acc[TPW_][8];
#pragma unroll
  for (int t = 0; t < TPW_; ++t)
#pragma unroll
    for (int r = 0; r < 8; ++r) acc[t][r] = 0.f;

  // u[b][n][i][l]: lane supplies A rows M=l15, K-pair = half*2 + {0,1}
  const float* ubase0 = u + ((size_t)l15 * NUM_ + n) * IN_N_ * IN_L_ + half * 2;
  // w[n][i][col][l]
  const float* wbase0 =
      w + (((size_t)n * IN_N_) * COLS_ + (size_t)wv * CPW_ + l15) * IN_L_ + half * 2;

#pragma unroll 1
  for (int ii = 0; ii < ICHUNK_; ++ii) {
    const int i = i0 + ii;
    const float* ub = ubase0 + (size_t)i * IN_L_;
    const float* wb = wbase0 + (size_t)i * COLS_ * IN_L_;
    if (ii + 1 < ICHUNK_) __builtin_prefetch(wb + (size_t)COLS_ * IN_L_, 0, 0);

    v2f af[4];
#pragma unroll
    for (int kk = 0; kk < 4; ++kk) af[kk] = *(const v2f*)(ub + kk * 4);

    if (routing) {
      // logits[b][o] = sum_v u_hat[b, o*32+v] * S[b][n][o][v]; tile pair (2t,2t+1) = one o
#pragma unroll
      for (int pr = 0; pr < TPW_ / 2; ++pr) {
        v8f d0 = mm16(wb + (size_t)(2 * pr) * 16 * IN_L_, af, bv[2 * pr]);
        v8f d1 = mm16(wb + (size_t)(2 * pr + 1) * 16 * IN_L_, af, bv[2 * pr + 1]);
        float p[8];
#pragma unroll
        for (int r = 0; r < 8; ++r) {
          int b = r + 8 * half;
          p[r] = d0[r] * S[sb[2 * pr] + b * BSTR_] + d1[r] * S[sb[2 * pr + 1] + b * BSTR_];
        }
#pragma unroll
        for (int m = 1; m < 16; m <<= 1)
#pragma unroll
          for (int r = 0; r < 8; ++r) p[r] += __shfl_xor(p[r], m, 32);
        if (l15 == 0) {
          int o = wv * (TPW_ / 2) + pr;
#pragma unroll
          for (int r = 0; r < 8; ++r) lg[r + 8 * half][o] = p[r];
        }
      }
      __syncthreads();
      if (tid < B_) {  // softmax over out_n, one batch row per thread
        float mx = -3.4e38f;
        for (int o = 0; o < OUT_N_; ++o) mx = fmaxf(mx, lg[tid][o]);
        float s = 0.f;
        for (int o = 0; o < OUT_N_; ++o) {
          float e = __expf(lg[tid][o] - mx);
          cc[tid][o] = e;
          s += e;
        }
        float inv = 1.f / s;
        for (int o = 0; o < OUT_N_; ++o) cc[tid][o] *= inv;
      }
      __syncthreads();
    }

    // accumulate cc * u_hat (recompute tiles; weights are L2-hot from phase 1)
#pragma unroll
    for (int t = 0; t < TPW_; ++t) {
      v8f d = mm16(wb + (size_t)t * 16 * IN_L_, af, bv[t]);
      int o = wv * (TPW_ / 2) + (t >> 1);
      if (routing) {
#pragma unroll
        for (int r = 0; r < 8; ++r) acc[t][r] += cc[r + 8 * half][o] * d[r];
      } else {
        float c = cvec[i * OUT_N_ + o];
#pragma unroll
        for (int r = 0; r < 8; ++r) acc[t][r] += c * d[r];
      }
    }
  }

  // block-level pre-reduced partials -> global vraw[b][n][o][v]
#pragma unroll
  for (int t = 0; t < TPW_; ++t)
#pragma unroll
    for (int r = 0; r < 8; ++r)
      unsafeAtomicAdd(&vraw[sb[t] + (r + 8 * half) * BSTR_], acc[t][r]);
}

// S += squash(vraw)  (or out = squash(vraw) on the final call); zero vraw for next pass.
__global__ __launch_bounds__(256) void caps_squash(float* __restrict__ vraw,
                                                   float* __restrict__ S,
                                                   float* __restrict__ out, int final_) {
  int row = blockIdx.x * 8 + (threadIdx.x >> 5);  // row over [b][n][o]
  int v = threadIdx.x & 31;
  int idx = row * OUT_L_ + v;
  float x = vraw[idx];
  float s = x * x;
#pragma unroll
  for (int m = 1; m < 32; m <<= 1) s += __shfl_xor(s, m, 32);
  float nrm = sqrtf(s);
  float n2 = nrm * nrm;
  float y = (n2 / (1.f + n2)) * x / (nrm + EPS_);
  if (final_) out[idx] = y;
  else S[idx] += y;
  vraw[idx] = 0.f;
}

__global__ void caps_zero(float* __restrict__ p, int nfloats) {
  int i = blockIdx.x * blockDim.x + threadIdx.x;
  if (i < nfloats) p[i] = 0.f;
}

extern "C" void kernel_launch(void* const* d_in, const int* in_sizes, int n_in, void* d_out,
                              int out_size, void* d_ws, size_t ws_size, hipStream_t stream) {
  const float* u    = (const float*)d_in[0];
  const float* w    = (const float*)d_in[1];
  const float* bias = (const float*)d_in[2];
  const float* cvec = (const float*)d_in[3];
  float* out  = (float*)d_out;
  float* vraw = (float*)d_ws;                          // 65536 floats
  float* S    = vraw + B_ * NUM_ * OUT_N_ * OUT_L_;    // 65536 floats

  int ztot = 2 * B_ * NUM_ * OUT_N_ * OUT_L_;
  caps_zero<<<(ztot + 255) / 256, 256, 0, stream>>>(vraw, ztot);

  dim3 pg(NUM_ * CHUNKS_), pb(512);                    // 288 blocks x 16 waves
  dim3 sg((B_ * NUM_ * OUT_N_) / 8), sbk(256);         // 2048 squash rows

  caps_pass<<<pg, pb, 0, stream>>>(u, w, bias, cvec, vraw, S, 0);   // v1 (cc = c)
  caps_squash<<<sg, sbk, 0, stream>>>(vraw, S, out, 0);             // S = v1
  caps_pass<<<pg, pb, 0, stream>>>(u, w, bias, cvec, vraw, S, 1);   // cc1 = softmax(ud.S)
  caps_squash<<<sg, sbk, 0, stream>>>(vraw, S, out, 0);             // S += v2
  caps_pass<<<pg, pb, 0, stream>>>(u, w, bias, cvec, vraw, S, 1);   // cc2
  caps_squash<<<sg, sbk, 0, stream>>>(vraw, S, out, 1);             // out = squash(.)
}